// BaseGraphConv_70153995813010
// MI455X (gfx1250) — compile-verified
//
#include <hip/hip_runtime.h>

#define NNODES 100000
#define NEDGES 1600000
#define FDIM   64
#define LDS_STRIDE 68   // pad 64 -> 68: float4-aligned rows, bank-swizzled fragment reads

typedef float v2f __attribute__((ext_vector_type(2)));
typedef float v8f __attribute__((ext_vector_type(8)));

// ---------------------------------------------------------------------------
// Zero the aggregation buffer (float4 stores, fully coalesced).
// ---------------------------------------------------------------------------
__global__ void zero_kernel(float4* __restrict__ p, int n4) {
    int i = blockIdx.x * blockDim.x + threadIdx.x;
    if (i < n4) p[i] = make_float4(0.f, 0.f, 0.f, 0.f);
}

// ---------------------------------------------------------------------------
// Edge scatter: agg[dst] += ew * h[src].  16 threads per edge, each owning a
// float4 feature chunk. Gather is an L2 hit (h = 25.6MB << 192MB L2);
// scatter uses hardware f32 atomic adds.
// ---------------------------------------------------------------------------
__global__ void scatter_kernel(const float* __restrict__ h,
                               const long long* __restrict__ eidx,
                               const float* __restrict__ ew,
                               float* __restrict__ agg) {
    int gid = blockIdx.x * blockDim.x + threadIdx.x;   // E*16 threads total
    int e   = gid >> 4;
    int fg  = (gid & 15) << 2;                          // feature chunk base
    if (e >= NEDGES) return;
    int src = (int)eidx[e];
    int dst = (int)eidx[NEDGES + e];
    float w = ew[e];
    const float4 v = *(const float4*)(h + (size_t)src * FDIM + fg);
    float* o = agg + (size_t)dst * FDIM + fg;
    unsafeAtomicAdd(o + 0, v.x * w);
    unsafeAtomicAdd(o + 1, v.y * w);
    unsafeAtomicAdd(o + 2, v.z * w);
    unsafeAtomicAdd(o + 3, v.w * w);
}

// ---------------------------------------------------------------------------
// Fused GraphConv tail: out = relu(agg @ Wrel^T + bias + h @ Wroot^T)
// One block = 16 output rows; wave w (of 4) owns 16 output columns.
// FP32 WMMA 16x16x4 chained over K=64 for BOTH matmuls into one accumulator.
// Safe for out == h (each block reads only its own rows before storing).
// ---------------------------------------------------------------------------
__global__ void __launch_bounds__(128)
graphconv_wmma_kernel(const float* __restrict__ A,      // agg  [N,64]
                      const float* __restrict__ Hm,     // h    [N,64]
                      const float* __restrict__ Wrel,   // [64,64] row-major [j][k]
                      const float* __restrict__ Wroot,  // [64,64]
                      const float* __restrict__ bias,   // [64]
                      float* __restrict__ out) {        // [N,64]
    __shared__ float ldsA[16 * LDS_STRIDE];
    __shared__ float ldsH[16 * LDS_STRIDE];

    const int tid  = threadIdx.x;
    const int row0 = blockIdx.x << 4;

    // Cooperative stage of the 16x64 A and H tiles (256 float4 each).
    for (int q = tid; q < 256; q += 128) {
        int r = q >> 4;
        int c = (q & 15) << 2;
        const float4 va = *(const float4*)(A  + (size_t)(row0 + r) * FDIM + c);
        const float4 vh = *(const float4*)(Hm + (size_t)(row0 + r) * FDIM + c);
        *(float4*)(&ldsA[r * LDS_STRIDE + c]) = va;
        *(float4*)(&ldsH[r * LDS_STRIDE + c]) = vh;
    }
    __syncthreads();

    const int lane = tid & 31;
    const int wid  = tid >> 5;          // column tile 0..3
    const int kh   = lane >> 4;         // lane half: selects K pair
    const int m    = lane & 15;         // A row / B column within tile
    const int nn   = (wid << 4) + m;    // global output column

    // Preload B fragments for both weight matrices (register-resident,
    // broadcast-hot in L1/L2 across all blocks).
    float br0[16], br1[16], bt0[16], bt1[16];
#pragma unroll
    for (int kc = 0; kc < 16; ++kc) {
        int k = (kc << 2) + (kh << 1);
        const float2 wr = *(const float2*)(Wrel  + nn * FDIM + k);
        const float2 wt = *(const float2*)(Wroot + nn * FDIM + k);
        br0[kc] = wr.x; br1[kc] = wr.y;
        bt0[kc] = wt.x; bt1[kc] = wt.y;
    }

    v8f acc = {};
#pragma unroll
    for (int kc = 0; kc < 16; ++kc) {
        int k = (kc << 2) + (kh << 1);
        // A-matrix fragment layout: lanes 0-15 hold K=kbase+{0,1},
        // lanes 16-31 hold K=kbase+{2,3}, M = lane&15.
        const float2 a2 = *(const float2*)(&ldsA[m * LDS_STRIDE + k]);
        const float2 h2 = *(const float2*)(&ldsH[m * LDS_STRIDE + k]);
        v2f af;  af.x  = a2.x; af.y  = a2.y;
        v2f hf;  hf.x  = h2.x; hf.y  = h2.y;
        v2f brf; brf.x = br0[kc]; brf.y = br1[kc];
        v2f btf; btf.x = bt0[kc]; btf.y = bt1[kc];
        acc = __builtin_amdgcn_wmma_f32_16x16x4_f32(
            false, af, false, brf, (short)0, acc, false, false);
        acc = __builtin_amdgcn_wmma_f32_16x16x4_f32(
            false, hf, false, btf, (short)0, acc, false, false);
    }

    // C/D layout: VGPR r, lanes 0-15 -> M=r, lanes 16-31 -> M=r+8; N = lane&15.
    const float bv = bias[nn];
#pragma unroll
    for (int r = 0; r < 8; ++r) {
        float v = acc[r] + bv;
        v = v > 0.f ? v : 0.f;
        out[(size_t)(row0 + r + (kh << 3)) * FDIM + nn] = v;
    }
}

// ---------------------------------------------------------------------------
extern "C" void kernel_launch(void* const* d_in, const int* in_sizes, int n_in,
                              void* d_out, int out_size, void* d_ws, size_t ws_size,
                              hipStream_t stream) {
    const float*     x       = (const float*)d_in[0];
    const long long* eidx    = (const long long*)d_in[1];   // int64 [2,E]
    const float*     ew      = (const float*)d_in[2];
    const float*     w1_rel  = (const float*)d_in[3];
    const float*     b1      = (const float*)d_in[4];
    const float*     w1_root = (const float*)d_in[5];
    const float*     w2_rel  = (const float*)d_in[6];
    const float*     b2      = (const float*)d_in[7];
    const float*     w2_root = (const float*)d_in[8];
    float* out = (float*)d_out;
    float* agg = (float*)d_ws;                              // N*64 f32 = 25.6 MB

    const int n4 = (NNODES * FDIM) / 4;
    dim3 zgrid((n4 + 255) / 256);
    dim3 sgrid((NEDGES * 16) / 256);   // 16 threads per edge
    dim3 ggrid(NNODES / 16);           // 100000 = 6250 * 16 exactly

    // Layer 1
    zero_kernel<<<zgrid, 256, 0, stream>>>((float4*)agg, n4);
    scatter_kernel<<<sgrid, 256, 0, stream>>>(x, eidx, ew, agg);
    graphconv_wmma_kernel<<<ggrid, 128, 0, stream>>>(agg, x, w1_rel, w1_root, b1, out);

    // Layer 2 (h1 lives in d_out; WMMA kernel reads/writes it block-locally)
    zero_kernel<<<zgrid, 256, 0, stream>>>((float4*)agg, n4);
    scatter_kernel<<<sgrid, 256, 0, stream>>>(out, eidx, ew, agg);
    graphconv_wmma_kernel<<<ggrid, 128, 0, stream>>>(agg, out, w2_rel, w2_root, b2, out);
}